// ProtoSimLoss_intra_calss_65687229826064
// MI455X (gfx1250) — compile-verified
//
#include <hip/hip_runtime.h>
#include <hip/hip_bf16.h>

#define N_SUBJECTS 16
#define N_LABELS 8
#define KGRP 128            // N_SUBJECTS * N_LABELS
#define BB 16384            // B
#define DD 1024             // D
#define BC 32768            // B*C
#define DENS_TEMP 0.1f

#define KCH 128             // K-chunk staged in LDS per pipeline stage
#define NCH (DD / KCH)      // 8 chunks
#define APAD 132            // padded LDS row stride (floats): 132 % 64 banks = 4 -> conflict-free column reads

typedef __attribute__((ext_vector_type(2))) float v2f;
typedef __attribute__((ext_vector_type(8))) float v8f;
typedef __attribute__((ext_vector_type(4))) int   v4i;

#if defined(__gfx1250__) && __has_builtin(__builtin_amdgcn_global_load_async_to_lds_b128)
#define USE_ASYNC_LDS 1
#else
#define USE_ASYNC_LDS 0
#endif

// ---------------- K1: gid = subject*8 + labels ----------------
__global__ __launch_bounds__(256) void k1_gid(const int* __restrict__ subject,
                                              const int* __restrict__ labels,
                                              int* __restrict__ gid) {
    int r = blockIdx.x * 256 + threadIdx.x;
    if (r < BB) gid[r] = subject[r] * N_LABELS + labels[r];
}

// ---------------- K2: centroid sums + exact counts ----------------
// grid = (KGRP, 4); each block owns (group g, 256-dim chunk). Register accumulation,
// no atomics; each row of X read by exactly one group -> X read once overall.
__global__ __launch_bounds__(256) void k2_centroid_sum(const float* __restrict__ X,
                                                       const int* __restrict__ gid,
                                                       float* __restrict__ centroid,
                                                       float* __restrict__ counts) {
    const int g = blockIdx.x;
    const int d = blockIdx.y * 256 + threadIdx.x;
    float acc = 0.f;
    int cnt = 0;
    for (int r = 0; r < BB; ++r) {
        if (gid[r] == g) {
            const float* p = X + (size_t)r * 2 * DD;   // X[r][0][*], X[r][1][*]
            acc += p[d] + p[DD + d];
            ++cnt;
        }
    }
    centroid[(size_t)g * DD + d] = acc;
    if (blockIdx.y == 0 && threadIdx.x == 0) counts[g] = (float)(2 * cnt);
}

// ---------------- K2b: centroid /= counts ----------------
__global__ __launch_bounds__(256) void k2b_div(float* __restrict__ centroid,
                                               const float* __restrict__ counts) {
    int i = blockIdx.x * 256 + threadIdx.x;   // 0 .. 128*1024-1
    centroid[i] = centroid[i] / counts[i >> 10];
}

// ---------------- K3: rowdist[row] = sumsq^(1/4) = sqrt(||feat - cen||) ----------------
__global__ __launch_bounds__(128) void k3_rowdist(const float* __restrict__ X,
                                                  const int* __restrict__ gid,
                                                  const float* __restrict__ centroid,
                                                  float* __restrict__ rowdist) {
    const int row = blockIdx.x;                       // 0..BC-1
    const int base = (row < BB) ? row : row - BB;
    const int ch = (row < BB) ? 0 : 1;
    const float* f = X + ((size_t)base * 2 + ch) * DD;
    const float* cen = centroid + (size_t)gid[base] * DD;
    float ss = 0.f;
    for (int d = threadIdx.x; d < DD; d += 128) {
        float df = f[d] - cen[d];
        ss += df * df;
    }
    __shared__ float red[128];
    red[threadIdx.x] = ss;
    __syncthreads();
    for (int s = 64; s > 0; s >>= 1) {
        if (threadIdx.x < s) red[threadIdx.x] += red[threadIdx.x + s];
        __syncthreads();
    }
    if (threadIdx.x == 0) rowdist[row] = sqrtf(sqrtf(red[0]));
}

// ---------------- K4a: per-group mean of rowdist ----------------
__global__ __launch_bounds__(256) void k4a_group_mean(const int* __restrict__ gid,
                                                      const float* __restrict__ rowdist,
                                                      const float* __restrict__ counts,
                                                      float* __restrict__ mean_sqrt) {
    const int g = blockIdx.x;
    float s = 0.f;
    for (int r = threadIdx.x; r < BC; r += 256) {
        int base = (r < BB) ? r : r - BB;
        if (gid[base] == g) s += rowdist[r];
    }
    __shared__ float red[256];
    red[threadIdx.x] = s;
    __syncthreads();
    for (int st = 128; st > 0; st >>= 1) {
        if (threadIdx.x < st) red[threadIdx.x] += red[threadIdx.x + st];
        __syncthreads();
    }
    if (threadIdx.x == 0) mean_sqrt[g] = red[0] / counts[g];
}

// ---------------- K4b: density (quantile/clip/normalize), one block of 128 ----------------
__global__ __launch_bounds__(128) void k4b_density(const float* __restrict__ mean_sqrt,
                                                   const float* __restrict__ counts,
                                                   float* __restrict__ density) {
    __shared__ float sd[KGRP];
    __shared__ float smax[KGRP];
    __shared__ float sq[5];
    const int g = threadIdx.x;
    float c = counts[g];
    float d = mean_sqrt[g] / logf(c + 10.f);
    smax[g] = d;
    __syncthreads();
    for (int s = 64; s > 0; s >>= 1) {
        if (g < s) smax[g] = fmaxf(smax[g], smax[g + s]);
        __syncthreads();
    }
    float dmax = smax[0];
    __syncthreads();
    d = (c <= 1.f) ? dmax : d;
    sd[g] = d;
    __syncthreads();
    // exact rank with index tiebreak -> sorted-order selection (jnp.quantile, linear interp)
    int rank = 0;
    for (int j = 0; j < KGRP; ++j) {
        float vj = sd[j];
        if (vj < d || (vj == d && j < g)) ++rank;
    }
    if (rank == 12)  sq[0] = d;   // (K-1)*0.1 = 12.7
    if (rank == 13)  sq[1] = d;
    if (rank == 114) sq[2] = d;   // (K-1)*0.9 = 114.3
    if (rank == 115) sq[3] = d;
    __syncthreads();
    float q10 = sq[0] + 0.7f * (sq[1] - sq[0]);
    float q90 = sq[2] + 0.3f * (sq[3] - sq[2]);
    d = fminf(fmaxf(d, q10), q90);
    sd[g] = d;
    __syncthreads();
    if (g == 0) {
        float s = 0.f;
        for (int j = 0; j < KGRP; ++j) s += sd[j];   // fixed order -> deterministic
        sq[4] = s * (1.f / (float)KGRP);
    }
    __syncthreads();
    density[g] = DENS_TEMP * d / sq[4];
}

// ---------------- K5: async-pipelined WMMA GEMM (fp32) + masked log-softmax loss ----------------
// grid = BC/16 = 2048 blocks; block = 256 threads = 8 waves.
// A tile (16 rows of feat) is staged into LDS in 8 double-buffered K-chunks via
// GLOBAL_LOAD_ASYNC_TO_LDS_B128 (ASYNCcnt); wave w computes cols [16w,16w+16) with
// v_wmma_f32_16x16x4_f32, A from LDS (conflict-free padded), B (centroid) from L2.
__global__ __launch_bounds__(256) void k5_gemm_loss(const float* __restrict__ X,
                                                    const int* __restrict__ subject,
                                                    const int* __restrict__ labels,
                                                    const float* __restrict__ centroid,
                                                    const float* __restrict__ density,
                                                    float* __restrict__ blockloss) {
    __shared__ float s_a[2][16 * APAD];  // 2 x 8448 B double-buffered A chunk
    __shared__ float s_sim[16 * KGRP];   // 8 KB
    __shared__ float s_loss[16];

    const int tid  = threadIdx.x;
    const int lane = tid & 31;
    const int wave = tid >> 5;
    const int l16  = lane & 15;
    const int lh   = lane >> 4;          // 0 or 1
    const int row0 = blockIdx.x * 16;
    const int col0 = wave * 16;

    // --- copy-stage mapping: 512 B128 segments per chunk (16 rows x 32 segs), 2 per thread
    const int s0 = tid, s1 = tid + 256;
    const int r0 = s0 >> 5, o0 = s0 & 31;     // row in tile, 16B-segment in chunk
    const int r1 = s1 >> 5, o1 = s1 & 31;
    const int gr0 = row0 + r0, gr1 = row0 + r1;
    const float* gp0 = X + ((gr0 < BB) ? ((size_t)gr0 * 2) : ((size_t)(gr0 - BB) * 2 + 1)) * DD + o0 * 4;
    const float* gp1 = X + ((gr1 < BB) ? ((size_t)gr1 * 2) : ((size_t)(gr1 - BB) * 2 + 1)) * DD + o1 * 4;
    float* lp0 = &s_a[0][r0 * APAD + o0 * 4];
    float* lp1 = &s_a[0][r1 * APAD + o1 * 4];
    const int lbufstride = 16 * APAD;

    // --- compute-stage A/B sources
    const float* brow = centroid + (size_t)(col0 + l16) * DD;   // B = centroid^T, K-major per lane
    const float* Abase = &s_a[0][l16 * APAD + 2 * lh];

    auto issue_chunk = [&](int c) {
        const int buf = c & 1;
        const float* g0 = gp0 + c * KCH;
        const float* g1 = gp1 + c * KCH;
        float* l0 = lp0 + buf * lbufstride;
        float* l1 = lp1 + buf * lbufstride;
#if USE_ASYNC_LDS
        __builtin_amdgcn_global_load_async_to_lds_b128((v4i*)g0, (v4i*)l0, 0, 0);
        __builtin_amdgcn_global_load_async_to_lds_b128((v4i*)g1, (v4i*)l1, 0, 0);
#else
        *(float4*)l0 = *(const float4*)g0;
        *(float4*)l1 = *(const float4*)g1;
#endif
    };

    issue_chunk(0);
    v8f acc = {};
    for (int c = 0; c < NCH; ++c) {
#if USE_ASYNC_LDS
        if (c + 1 < NCH) {
            issue_chunk(c + 1);                               // keep next chunk in flight
            asm volatile("s_wait_asynccnt 0x2" ::: "memory"); // chunk c done (in-order), c+1 pending
        } else {
            asm volatile("s_wait_asynccnt 0x0" ::: "memory");
        }
#else
        if (c + 1 < NCH) issue_chunk(c + 1);
#endif
        __syncthreads();   // all waves' chunk-c LDS writes visible

        const float* Ab = Abase + (c & 1) * lbufstride;       // lane row l16, K-offset 2*lh folded in
        const float* Bb = brow + c * KCH + 2 * lh;
        __builtin_prefetch(Bb + KCH, 0, 3);                   // global_prefetch_b8: next B chunk
        #pragma unroll 4
        for (int kl = 0; kl < KCH; kl += 4) {
            v2f a, b;
            a.x = Ab[kl];        // A 16x4 layout: VGPR0 = K=k0 / k0+2, VGPR1 = K=k0+1 / k0+3
            a.y = Ab[kl + 1];
            b.x = Bb[kl];
            b.y = Bb[kl + 1];
            acc = __builtin_amdgcn_wmma_f32_16x16x4_f32(
                /*neg_a=*/false, a, /*neg_b=*/false, b,
                /*c_mod=*/(short)0, acc, /*reuse_a=*/false, /*reuse_b=*/false);
        }
        __syncthreads();   // everyone done reading buf[c&1] before chunk c+2 overwrites it
    }

    // Epilogue: sim /= density[col]; C/D layout: lane L -> col = L%16, M = v + 8*(L/16)
    const float inv_d = 1.0f / density[col0 + l16];
    #pragma unroll
    for (int v = 0; v < 8; ++v) {
        int m = v + 8 * lh;
        s_sim[m * KGRP + col0 + l16] = acc[v] * inv_d;
    }
    __syncthreads();

    // Per-row masked log-softmax loss (threads 0..15, one per row).
    if (tid < 16) {
        const int grow = row0 + tid;
        const int base = (grow < BB) ? grow : grow - BB;
        const int sr = subject[base];
        const int lr = labels[base];
        const float* srow = s_sim + tid * KGRP;
        float mx = -3.402823466e38f;
        for (int g = 0; g < KGRP; ++g) mx = fmaxf(mx, srow[g]);
        float sumexp = 0.f, possum = 0.f;
        int npos = 0;
        for (int g = 0; g < KGRP; ++g) {
            float s = srow[g] - mx;
            bool msk = ((g >> 3) != sr) && ((g & 7) == lr);
            float p = msk ? s : 0.f;       // pos = sim_shifted * mask
            sumexp += expf(p);
            if (msk) { possum += p; ++npos; }
        }
        float lse = logf(sumexp);
        s_loss[tid] = -(possum - (float)npos * lse) / (float)npos;
    }
    __syncthreads();
    if (tid == 0) {
        float s = 0.f;
        for (int t = 0; t < 16; ++t) s += s_loss[t];  // fixed order
        blockloss[blockIdx.x] = s;
    }
}

// ---------------- K6: final deterministic mean ----------------
__global__ __launch_bounds__(256) void k6_final(const float* __restrict__ blockloss,
                                                float* __restrict__ out) {
    __shared__ float red[256];
    float s = 0.f;
    for (int i = threadIdx.x; i < BC / 16; i += 256) s += blockloss[i];
    red[threadIdx.x] = s;
    __syncthreads();
    for (int st = 128; st > 0; st >>= 1) {
        if (threadIdx.x < st) red[threadIdx.x] += red[threadIdx.x + st];
        __syncthreads();
    }
    if (threadIdx.x == 0) out[0] = red[0] * (1.f / (float)BC);
}

extern "C" void kernel_launch(void* const* d_in, const int* in_sizes, int n_in,
                              void* d_out, int out_size, void* d_ws, size_t ws_size,
                              hipStream_t stream) {
    const float* X       = (const float*)d_in[0];   // (B, C, D) fp32
    const int*   subject = (const int*)d_in[1];     // (B,)
    const int*   labels  = (const int*)d_in[2];     // (B,)
    float* out = (float*)d_out;

    // workspace carve (all fully written before read; no zero-init needed)
    float* ws       = (float*)d_ws;
    int*   gid      = (int*)ws;                   // 16384 ints
    float* centroid = ws + BB;                    // 128*1024
    float* counts   = centroid + KGRP * DD;       // 128
    float* rowdist  = counts + KGRP;              // 32768
    float* meansq   = rowdist + BC;               // 128
    float* density  = meansq + KGRP;              // 128
    float* blkloss  = density + KGRP;             // 2048

    k1_gid<<<BB / 256, 256, 0, stream>>>(subject, labels, gid);
    k2_centroid_sum<<<dim3(KGRP, DD / 256), 256, 0, stream>>>(X, gid, centroid, counts);
    k2b_div<<<(KGRP * DD) / 256, 256, 0, stream>>>(centroid, counts);
    k3_rowdist<<<BC, 128, 0, stream>>>(X, gid, centroid, rowdist);
    k4a_group_mean<<<KGRP, 256, 0, stream>>>(gid, rowdist, counts, meansq);
    k4b_density<<<1, KGRP, 0, stream>>>(meansq, counts, density);
    k5_gemm_loss<<<BC / 16, 256, 0, stream>>>(X, subject, labels, centroid, density, blkloss);
    k6_final<<<1, 256, 0, stream>>>(blkloss, out);
}